// custom_GATConv_77816217469102
// MI455X (gfx1250) — compile-verified
//
#include <hip/hip_runtime.h>
#include <hip/hip_bf16.h>

// ---------------------------------------------------------------------------
// GATConv for MI455X (gfx1250, wave32).
//   h = feat @ W_fc^T  via V_WMMA_F32_16X16X4_F32 (exact fp32 matmul)
//   edge softmax + scatter phases run out of L2 (h = 51.2MB < 192MB L2).
// Problem constants from the reference: IN=256, H=4, D=32 -> HD=128.
// ---------------------------------------------------------------------------

typedef __attribute__((ext_vector_type(2))) float v2f;
typedef __attribute__((ext_vector_type(8))) float v8f;

#define IN_DIM     256
#define HD         128       // H*D
#define NHEAD      4
#define DDIM       32
#define NEG_SLOPE  0.2f
#define TILE_ROWS  32
#define LDS_STRIDE 260       // 256 + 4 pad -> conflict-free b64 LDS reads

// ---------------------------------------------------------------------------
// Kernel 1: h[N,128] = feat[N,256] @ W_fc[128,256]^T
// Block = 256 threads = 8 waves. Block tile: 32 rows x 128 cols.
// Wave w owns columns [16w, 16w+16) with two 16x16 f32 accumulators
// (rows 0-15 and 16-31 of the tile). A staged in LDS, B (W_fc, 128KB)
// served from cache.
//
// WMMA f32 16x16x4 fragment layouts (ISA 7.12.2):
//   A (16x4):  lane m = lane&15, kpair = 2*(lane>>4); v2f = A[m, k+kp..kp+1]
//   B (4x16):  lane n = lane&15, kpair = 2*(lane>>4); v2f = B[kp..kp+1, n]
//   C/D (16x16): col n = lane&15, row = vgpr + 8*(lane>>4)
// ---------------------------------------------------------------------------
__global__ __launch_bounds__(256) void gat_gemm_wmma(
    const float* __restrict__ feat, const float* __restrict__ Wfc,
    float* __restrict__ h, int N)
{
    __shared__ float As[TILE_ROWS * LDS_STRIDE];   // 33.3 KB of 320 KB WGP LDS

    const int row0 = blockIdx.x * TILE_ROWS;
    const int tid  = threadIdx.x;

    // Cooperative load: 32 rows x 64 float4 = 2048 float4, 8 per thread.
    for (int i = tid; i < TILE_ROWS * 64; i += 256) {
        const int r  = i >> 6;
        const int c4 = i & 63;
        const int gr = row0 + r;
        float4 v = make_float4(0.f, 0.f, 0.f, 0.f);
        if (gr < N) v = ((const float4*)(feat + (size_t)gr * IN_DIM))[c4];
        float* p = &As[r * LDS_STRIDE + c4 * 4];
        p[0] = v.x; p[1] = v.y; p[2] = v.z; p[3] = v.w;
    }
    __syncthreads();

    const int wave = tid >> 5;          // 0..7 -> 16-column strip
    const int lane = tid & 31;
    const int col0 = wave * 16;
    const int l15  = lane & 15;         // = m for A, = n for B/D
    const int kb   = (lane >> 4) * 2;   // K sub-pair within step-of-4

    v8f c0 = {};                        // rows row0+0..15
    v8f c1 = {};                        // rows row0+16..31

    const float* wrow = Wfc + (size_t)(col0 + l15) * IN_DIM;  // B column n
    const float* arow = &As[l15 * LDS_STRIDE];

    for (int k0 = 0; k0 < IN_DIM; k0 += 4) {
        v2f b;
        b.x = wrow[k0 + kb];
        b.y = wrow[k0 + kb + 1];

        v2f a0, a1;
        const float* ap0 = arow + k0 + kb;
        a0.x = ap0[0];  a0.y = ap0[1];
        const float* ap1 = ap0 + 16 * LDS_STRIDE;
        a1.x = ap1[0];  a1.y = ap1[1];

        c0 = __builtin_amdgcn_wmma_f32_16x16x4_f32(
                 false, a0, false, b, (short)0, c0, false, false);
        c1 = __builtin_amdgcn_wmma_f32_16x16x4_f32(
                 false, a1, false, b, (short)0, c1, false, false);
    }

    const int mbase = (lane >> 4) * 8;
    const int col   = col0 + l15;
    #pragma unroll
    for (int v = 0; v < 8; ++v) {
        const int r0 = row0 + mbase + v;
        if (r0 < N)      h[(size_t)r0 * HD + col]        = c0[v];
        if (r0 + 16 < N) h[(size_t)(r0 + 16) * HD + col] = c1[v];
    }
}

// ---------------------------------------------------------------------------
// Kernel 2: el[n,h] = dot(h[n,h,:], attn_l[h,:]);  er likewise.
// ---------------------------------------------------------------------------
__global__ void gat_el_er(const float* __restrict__ h,
                          const float* __restrict__ attn_l,
                          const float* __restrict__ attn_r,
                          float* __restrict__ el, float* __restrict__ er, int N)
{
    const int i = blockIdx.x * blockDim.x + threadIdx.x;    // over N*H
    if (i >= N * NHEAD) return;
    const int n = i >> 2, hh = i & 3;
    const float* hp = h + (size_t)n * HD + hh * DDIM;
    const float* al = attn_l + hh * DDIM;
    const float* ar = attn_r + hh * DDIM;
    float sl = 0.f, sr = 0.f;
    #pragma unroll
    for (int d = 0; d < DDIM; ++d) {
        const float x = hp[d];
        sl += x * al[d];
        sr += x * ar[d];
    }
    el[i] = sl;
    er[i] = sr;
}

// ---------------------------------------------------------------------------
// Kernel 3: init emax=-inf, denom=0, out=bias (broadcast).
// ---------------------------------------------------------------------------
__global__ void gat_init(float* __restrict__ emax, float* __restrict__ denom,
                         float* __restrict__ out, const float* __restrict__ bias,
                         int N)
{
    const int i = blockIdx.x * blockDim.x + threadIdx.x;    // over N*128
    if (i < N * NHEAD) {
        emax[i]  = -__builtin_inff();
        denom[i] = 0.f;
    }
    if (i < N * HD) out[i] = bias[i & (HD - 1)];
}

// Monotone float atomic-max: sign-split int/uint trick (init must be -inf,
// which it is). Correct under any interleaving; no CAS loop needed.
__device__ __forceinline__ void atomicMaxF32(float* addr, float v)
{
    if (v >= 0.f)
        atomicMax((int*)addr, __float_as_int(v));
    else
        atomicMin((unsigned int*)addr, __float_as_uint(v));
}

__device__ __forceinline__ float leaky(float v)
{
    return v > 0.f ? v : NEG_SLOPE * v;
}

// ---------------------------------------------------------------------------
// Kernel 4: segment max of e over dst (one thread per edge*head).
// ---------------------------------------------------------------------------
__global__ void gat_edge_max(const int* __restrict__ src, const int* __restrict__ dst,
                             const float* __restrict__ el, const float* __restrict__ er,
                             float* __restrict__ emax, int E)
{
    const int i = blockIdx.x * blockDim.x + threadIdx.x;    // over E*H
    if (i >= E * NHEAD) return;
    const int e = i >> 2, hh = i & 3;
    const int s = src[e], d = dst[e];
    const float v = leaky(el[s * NHEAD + hh] + er[d * NHEAD + hh]);
    atomicMaxF32(&emax[d * NHEAD + hh], v);
}

// ---------------------------------------------------------------------------
// Kernel 5: denom[dst] += exp(e - emax[dst]).
// ---------------------------------------------------------------------------
__global__ void gat_edge_sum(const int* __restrict__ src, const int* __restrict__ dst,
                             const float* __restrict__ el, const float* __restrict__ er,
                             const float* __restrict__ emax, float* __restrict__ denom,
                             int E)
{
    const int i = blockIdx.x * blockDim.x + threadIdx.x;    // over E*H
    if (i >= E * NHEAD) return;
    const int e = i >> 2, hh = i & 3;
    const int s = src[e], d = dst[e];
    const float v = leaky(el[s * NHEAD + hh] + er[d * NHEAD + hh]);
    atomicAdd(&denom[d * NHEAD + hh], __expf(v - emax[d * NHEAD + hh]));
}

// ---------------------------------------------------------------------------
// Kernel 6: out[dst] += h[src] * (a + h[dst]);  a = exp(e-emax)/denom.
// One thread per (edge, hd-element). Per-warp the head index is constant,
// so el/er/emax/denom loads broadcast; h gathers hit L2 (h fits in 192MB L2).
// ---------------------------------------------------------------------------
__global__ void gat_edge_aggr(const int* __restrict__ src, const int* __restrict__ dst,
                              const float* __restrict__ el, const float* __restrict__ er,
                              const float* __restrict__ emax, const float* __restrict__ denom,
                              const float* __restrict__ h, float* __restrict__ out, int E)
{
    const int i = blockIdx.x * blockDim.x + threadIdx.x;    // over E*128
    if (i >= E * HD) return;
    const int e  = i >> 7;
    const int j  = i & (HD - 1);
    const int hh = j >> 5;
    const int s  = src[e], d = dst[e];

    const float v = leaky(el[s * NHEAD + hh] + er[d * NHEAD + hh]);
    const float a = __expf(v - emax[d * NHEAD + hh]) / denom[d * NHEAD + hh];

    const float hs = h[(size_t)s * HD + j];
    const float hd = h[(size_t)d * HD + j];
    atomicAdd(&out[(size_t)d * HD + j], hs * (a + hd));
}

// ---------------------------------------------------------------------------
// Launcher. Workspace layout (fp32):
//   h[N*128] | el[N*4] | er[N*4] | emax[N*4] | denom[N*4]  (~57.6 MB)
// ---------------------------------------------------------------------------
extern "C" void kernel_launch(void* const* d_in, const int* in_sizes, int n_in,
                              void* d_out, int out_size, void* d_ws, size_t ws_size,
                              hipStream_t stream)
{
    const float* feat   = (const float*)d_in[0];
    const float* Wfc    = (const float*)d_in[1];
    const float* attn_l = (const float*)d_in[2];
    const float* attn_r = (const float*)d_in[3];
    const float* bias   = (const float*)d_in[4];
    const int*   src    = (const int*)d_in[5];
    const int*   dst    = (const int*)d_in[6];

    const int N = in_sizes[0] / IN_DIM;
    const int E = in_sizes[5];

    float* h     = (float*)d_ws;
    float* el    = h    + (size_t)N * HD;
    float* er    = el   + (size_t)N * NHEAD;
    float* emax  = er   + (size_t)N * NHEAD;
    float* denom = emax + (size_t)N * NHEAD;
    float* out   = (float*)d_out;

    const int gemm_blocks = (N + TILE_ROWS - 1) / TILE_ROWS;
    gat_gemm_wmma<<<gemm_blocks, 256, 0, stream>>>(feat, Wfc, h, N);

    gat_el_er<<<(N * NHEAD + 255) / 256, 256, 0, stream>>>(h, attn_l, attn_r, el, er, N);

    gat_init<<<(N * HD + 255) / 256, 256, 0, stream>>>(emax, denom, out, bias, N);

    gat_edge_max<<<(E * NHEAD + 255) / 256, 256, 0, stream>>>(src, dst, el, er, emax, E);

    gat_edge_sum<<<(E * NHEAD + 255) / 256, 256, 0, stream>>>(src, dst, el, er, emax, denom, E);

    gat_edge_aggr<<<(E * HD + 255) / 256, 256, 0, stream>>>(src, dst, el, er, emax, denom, h, out, E);
}